// PostProcess_16733192585466
// MI455X (gfx1250) — compile-verified
//
#include <hip/hip_runtime.h>
#include <stdint.h>

// ---------------- problem constants (from reference) ----------------
#define B_IMG      8
#define N_CAND     25200
#define NUM_CLS    80
#define NPAD       32768          // 2^15 >= N_CAND, bitonic-sortable
#define TOPK_K     2048
#define MAX_DET    300
#define CONF_THR   0.2f
#define IOU_THR    0.6f
#define MAX_WH_F   4096.0f

typedef __attribute__((ext_vector_type(2))) float v2f;
typedef __attribute__((ext_vector_type(8))) float v8f;
typedef int v2i __attribute__((vector_size(8)));   // matches builtin param type

#define AS1 __attribute__((address_space(1)))
#define AS3 __attribute__((address_space(3)))

#if defined(__gfx1250__) && __has_builtin(__builtin_amdgcn_global_load_async_to_lds_b64) && __has_builtin(__builtin_amdgcn_s_wait_asynccnt)
#define HAVE_ASYNC_LDS 1
#else
#define HAVE_ASYNC_LDS 0
#endif

#if defined(__gfx1250__) && __has_builtin(__builtin_amdgcn_wmma_f32_16x16x4_f32)
#define HAVE_WMMA_F32 1
#else
#define HAVE_WMMA_F32 0
#endif

__device__ __forceinline__ unsigned wave_ballot32(bool p) {
#if defined(__gfx1250__) && __has_builtin(__builtin_amdgcn_ballot_w32)
  return __builtin_amdgcn_ballot_w32(p);
#else
  return (unsigned)__ballot(p);
#endif
}

// Decode one candidate row (85 floats): xyxy box, best class, conf = max(obj*cls)
__device__ __forceinline__ void decode_cand(const float* __restrict__ p,
                                            float4& box, int& bestc, float& conf) {
  const float obj = p[4];
  float best = p[5] * obj;
  int bc = 0;
#pragma unroll 8
  for (int c = 1; c < NUM_CLS; ++c) {
    float v = p[5 + c] * obj;
    if (v > best) { best = v; bc = c; }
  }
  const float hx = 0.5f * p[2], hy = 0.5f * p[3];
  box = make_float4(p[0] - hx, p[1] - hy, p[0] + hx, p[1] + hy);
  bestc = bc;
  conf = best;
}

// ---------------- K1: decode -> sortable 64-bit keys ----------------
// key = (score_bits << 32) | ~idx   (descending score, ascending index on ties)
__global__ void __launch_bounds__(256)
decode_keys_kernel(const float* __restrict__ preds,
                   unsigned long long* __restrict__ keys) {
  const int gid = blockIdx.x * blockDim.x + threadIdx.x;   // B_IMG * NPAD threads
  const int img = gid >> 15;            // NPAD == 2^15
  const int idx = gid & (NPAD - 1);
  unsigned long long key = 0ull;
  if (idx < N_CAND) {
    const float* p = preds + ((size_t)img * N_CAND + idx) * 85;
    __builtin_prefetch(p + 85 * 64, 0, 0);   // global_prefetch_b8, speculative
    float4 box; int bc; float conf;
    decode_cand(p, box, bc, conf);
    const float score = (conf > CONF_THR) ? conf : 0.0f;
    key = ((unsigned long long)__float_as_uint(score) << 32)
        | (unsigned long long)(~(unsigned)idx);
  }
  keys[gid] = key;
}

// ---------------- K2: per-image full bitonic sort in 256KB LDS, gather top-K ----------
__global__ void __launch_bounds__(1024)
sort_gather_kernel(const unsigned long long* __restrict__ keys,
                   const float* __restrict__ preds,
                   float4* __restrict__ tbox, float4* __restrict__ tobox,
                   float* __restrict__ tarea, float* __restrict__ tscore,
                   float* __restrict__ tcls) {
  extern __shared__ unsigned long long smem[];   // NPAD u64 = 256 KB (CDNA5: 320KB/WGP)
  const int img = blockIdx.x;
  const int tid = threadIdx.x;
  const int nt  = blockDim.x;
  const unsigned long long* kg = keys + (size_t)img * NPAD;

#if HAVE_ASYNC_LDS
  for (int t = tid; t < NPAD; t += nt) {
    __builtin_amdgcn_global_load_async_to_lds_b64(
        (AS1 v2i*)(kg + t), (AS3 v2i*)(smem + t), 0, 0);
  }
  __builtin_amdgcn_s_wait_asynccnt(0);
#else
  for (int t = tid; t < NPAD; t += nt) smem[t] = kg[t];
#endif
  __syncthreads();

  // Bitonic sort, descending (largest key first).
  for (int k = 2; k <= NPAD; k <<= 1) {
    for (int j = k >> 1; j > 0; j >>= 1) {
      for (int t = tid; t < NPAD; t += nt) {
        const int ixj = t ^ j;
        if (ixj > t) {
          const unsigned long long a = smem[t], b = smem[ixj];
          const bool up = ((t & k) == 0);
          if (up ? (a < b) : (a > b)) { smem[t] = b; smem[ixj] = a; }
        }
      }
      __syncthreads();
    }
  }

  // Gather top-K (sorted), re-decode box/class from preds (saves workspace).
  for (int q = tid; q < TOPK_K; q += nt) {
    const unsigned long long key = smem[q];
    const float sc = __uint_as_float((unsigned)(key >> 32));
    unsigned idx = ~(unsigned)key;
    if (idx >= N_CAND) idx = 0;               // zero-key padding -> harmless row
    const float* p = preds + ((size_t)img * N_CAND + idx) * 85;
    float4 b; int bc; float conf;
    decode_cand(p, b, bc, conf);
    const float off = (float)bc * MAX_WH_F;
    const float4 obx = make_float4(b.x + off, b.y + off, b.z + off, b.w + off);
    const size_t o = (size_t)img * TOPK_K + q;
    tbox[o]  = b;
    tobox[o] = obx;
    // area computed from OFFSET boxes, matching reference rounding
    tarea[o] = (obx.z - obx.x) * (obx.w - obx.y);
    tscore[o] = sc;
    tcls[o]  = (float)bc;
  }
}

// ---------------- K3: pairwise IoU suppression bit-matrix (wave = 16x32 tile) ----------
// Union denominator area_i + area_j computed as rank-2 outer sum via
// v_wmma_f32_16x16x4_f32:  A rows = [area_i, 1, 0, 0],  B cols = [1, area_j, 0, 0].
__global__ void __launch_bounds__(256)
iou_mask_kernel(const float4* __restrict__ tobox,
                const float* __restrict__ tarea,
                unsigned* __restrict__ mask) {
  const int tid  = blockIdx.x * blockDim.x + threadIdx.x;
  const int wid  = tid >> 5;            // wave32
  const int lane = threadIdx.x & 31;
  const int img  = wid >> 13;           // 8192 waves per image
  const int rem  = wid & 8191;
  const int it   = rem >> 6;            // 0..127 i-tile (16 rows)
  const int jb   = rem & 63;            // 0..63 j-block (32 cols)
  const int i0   = it << 4;
  const int j0   = jb << 5;
  const float4* ob = tobox + (size_t)img * TOPK_K;
  const float*  ar = tarea + (size_t)img * TOPK_K;
  const int ln = lane & 15;
  const int g  = lane >> 4;

  float dsum0[8], dsum1[8];             // (M = r + 8g, N = ln) per WMMA C/D layout
#if HAVE_WMMA_F32
  {
    v2f a, b0, b1;
    // A 16x4 f32 layout: lanes 0-15 hold K0(V0),K1(V1) of row M=lane; lanes 16-31 hold K2,K3 (=0)
    a.x = (g == 0) ? ar[i0 + ln] : 0.0f;
    a.y = (g == 0) ? 1.0f : 0.0f;
    // B 4x16 f32 layout: lanes 0-15 hold K0(V0),K1(V1) of col N=lane; lanes 16-31 hold K2,K3 (=0)
    b0.x = (g == 0) ? 1.0f : 0.0f;
    b0.y = (g == 0) ? ar[j0 + ln] : 0.0f;
    b1.x = (g == 0) ? 1.0f : 0.0f;
    b1.y = (g == 0) ? ar[j0 + 16 + ln] : 0.0f;
    v8f z = {0.f, 0.f, 0.f, 0.f, 0.f, 0.f, 0.f, 0.f};
    v8f d0 = __builtin_amdgcn_wmma_f32_16x16x4_f32(false, a, false, b0, (short)0, z, false, false);
    v8f d1 = __builtin_amdgcn_wmma_f32_16x16x4_f32(false, a, false, b1, (short)0, z, false, false);
#pragma unroll
    for (int r = 0; r < 8; ++r) { dsum0[r] = d0[r]; dsum1[r] = d1[r]; }
  }
#else
#pragma unroll
  for (int r = 0; r < 8; ++r) {
    const float ai = ar[i0 + r + (g << 3)];
    dsum0[r] = ai + ar[j0 + ln];
    dsum1[r] = ai + ar[j0 + 16 + ln];
  }
#endif

  const int jA = j0 + ln, jB = j0 + 16 + ln;
  const float4 bA = ob[jA];
  const float4 bB = ob[jB];
  unsigned* mrow = mask + ((size_t)img * TOPK_K + i0) * 64 + jb;

#pragma unroll
  for (int r = 0; r < 8; ++r) {
    const int i = i0 + r + (g << 3);
    const float4 bi = ob[i];
    float iw0 = fmaxf(fminf(bi.z, bA.z) - fmaxf(bi.x, bA.x), 0.0f);
    float ih0 = fmaxf(fminf(bi.w, bA.w) - fmaxf(bi.y, bA.y), 0.0f);
    const float inter0 = iw0 * ih0;
    const float iou0 = inter0 / (dsum0[r] - inter0 + 1e-9f);
    const bool p0 = (iou0 > IOU_THR) && (jA > i);
    float iw1 = fmaxf(fminf(bi.z, bB.z) - fmaxf(bi.x, bB.x), 0.0f);
    float ih1 = fmaxf(fminf(bi.w, bB.w) - fmaxf(bi.y, bB.y), 0.0f);
    const float inter1 = iw1 * ih1;
    const float iou1 = inter1 / (dsum1[r] - inter1 + 1e-9f);
    const bool p1 = (iou1 > IOU_THR) && (jB > i);
    const unsigned bal0 = wave_ballot32(p0);
    const unsigned bal1 = wave_ballot32(p1);
    if (ln == 0) {
      const unsigned wbits = ((bal0 >> (g << 4)) & 0xFFFFu)
                           | (((bal1 >> (g << 4)) & 0xFFFFu) << 16);
      mrow[(size_t)(r + (g << 3)) * 64] = wbits;
    }
  }
}

// ---------------- K4: greedy sequential NMS over bitmask; emit top-300 ----------------
__global__ void __launch_bounds__(32)
nms_finalize_kernel(const unsigned* __restrict__ mask,
                    const float4* __restrict__ tbox,
                    const float* __restrict__ tscore,
                    const float* __restrict__ tcls,
                    float* __restrict__ out) {
  const int img  = blockIdx.x;
  const int lane = threadIdx.x;            // blockDim == 32 (one wave)
  const unsigned* mbase = mask + (size_t)img * TOPK_K * 64;
  const float*  sc = tscore + (size_t)img * TOPK_K;
  const float4* tb = tbox   + (size_t)img * TOPK_K;
  const float*  tc = tcls   + (size_t)img * TOPK_K;
  float* o = out + (size_t)img * MAX_DET * 6;

  unsigned w0 = 0u, w1 = 0u;               // lane owns suppression words lane, lane+32
  int count = 0;
  for (int i = 0; i < TOPK_K; ++i) {
    const float s = sc[i];                 // uniform broadcast load
    const int word  = i >> 5;
    const int owner = word & 31;
    const int hi    = word >> 5;
    const unsigned v0 = __shfl(w0, owner, 32);
    const unsigned v1 = __shfl(w1, owner, 32);
    const unsigned wv = hi ? v1 : v0;
    const bool sup = (wv >> (i & 31)) & 1u;
    if ((s > CONF_THR) && !sup) {          // uniform condition across the wave
      const unsigned* mrow = mbase + (size_t)i * 64;
      w0 |= mrow[lane];
      w1 |= mrow[lane + 32];
      if (count < MAX_DET && lane == 0) {
        const float4 b = tb[i];
        float* r = o + count * 6;
        r[0] = b.x; r[1] = b.y; r[2] = b.z; r[3] = b.w;
        r[4] = s;   r[5] = tc[i];
      }
      count++;
    }
  }
  const int start = ((count < MAX_DET) ? count : MAX_DET) * 6;
  for (int t = start + lane; t < MAX_DET * 6; t += 32) o[t] = 0.0f;
}

// ---------------- launch ----------------
extern "C" void kernel_launch(void* const* d_in, const int* in_sizes, int n_in,
                              void* d_out, int out_size, void* d_ws, size_t ws_size,
                              hipStream_t stream) {
  (void)in_sizes; (void)n_in; (void)out_size; (void)ws_size;
  const float* preds = (const float*)d_in[0];
  float* out = (float*)d_out;
  char* ws = (char*)d_ws;

  size_t off = 0;
  auto take = [&](size_t bytes) -> void* {
    void* p = ws + off;
    off += (bytes + 255) & ~(size_t)255;
    return p;
  };
  unsigned long long* keys = (unsigned long long*)take((size_t)B_IMG * NPAD * 8);          // 2 MB
  float4* tbox   = (float4*)take((size_t)B_IMG * TOPK_K * sizeof(float4));                 // 256 KB
  float4* tobox  = (float4*)take((size_t)B_IMG * TOPK_K * sizeof(float4));                 // 256 KB
  float*  tarea  = (float*) take((size_t)B_IMG * TOPK_K * sizeof(float));                  // 64 KB
  float*  tscore = (float*) take((size_t)B_IMG * TOPK_K * sizeof(float));                  // 64 KB
  float*  tcls   = (float*) take((size_t)B_IMG * TOPK_K * sizeof(float));                  // 64 KB
  unsigned* mask = (unsigned*)take((size_t)B_IMG * TOPK_K * 64 * sizeof(unsigned));        // 4 MB

  decode_keys_kernel<<<(B_IMG * NPAD) / 256, 256, 0, stream>>>(preds, keys);
  sort_gather_kernel<<<B_IMG, 1024, (size_t)NPAD * sizeof(unsigned long long), stream>>>(
      keys, preds, tbox, tobox, tarea, tscore, tcls);
  // 8 images * 128 i-tiles * 64 j-blocks = 65536 waves -> 8192 blocks of 8 waves
  iou_mask_kernel<<<(B_IMG * 128 * 64 * 32) / 256, 256, 0, stream>>>(tobox, tarea, mask);
  nms_finalize_kernel<<<B_IMG, 32, 0, stream>>>(mask, tbox, tscore, tcls, out);
}